// BEVFormerEncoder_51153060495941
// MI455X (gfx1250) — compile-verified
//
#include <hip/hip_runtime.h>
#include <hip/hip_bf16.h>
#include <math.h>

#define NCAM 6
#define DM 256
#define NH 8
#define HD 32
#define NLV 4
#define NPT 4
#define NZC 2
#define BEV_H 100
#define BEV_W 100
#define NQ (BEV_H * BEV_W)
#define DFF 1024

typedef __attribute__((ext_vector_type(2))) float v2f;
typedef __attribute__((ext_vector_type(8))) float v8f;

// ---------------------------------------------------------------------------
// Register-blocked fp32 WMMA GEMM: each wave computes a 64x32 macro tile as a
// 4x2 grid of 16x16 V_WMMA_F32_16X16X4_F32 accumulators (64 acc VGPRs -> no
// spills). Requires N % 32 == 0. Per K-step (4): 4 A b64 loads + 4 B b32
// loads feed 8 WMMAs -> ~10.7 FLOPs/byte of memory traffic.
// A frag layout: lanes 0-15 hold M=lr with K+0/K+1; lanes 16-31 K+2/K+3.
// B frag layout: lanes 0-15 hold N=lr with K+0/K+1; lanes 16-31 K+2/K+3.
// C/D: half 0 -> rows 0..7, half 1 -> rows 8..15, column = lane&15.
// ---------------------------------------------------------------------------
__global__ __launch_bounds__(256, 1) void gemm_wmma_f32_64x32(
    const float* __restrict__ A, const float* __restrict__ W,
    const float* __restrict__ bias, float* __restrict__ C,
    int M, int K, int N, int relu) {
  const int lane = threadIdx.x & 31;
  const int wave = threadIdx.x >> 5;
  const int ntc = N >> 5;                       // N / 32
  const int mtc = (M + 63) >> 6;                // ceil(M / 64)
  const long tile = (long)blockIdx.x * 8 + wave;
  if (tile >= (long)mtc * ntc) return;          // uniform per wave
  const int mt = (int)(tile / ntc);
  const int nt = (int)(tile % ntc);

  const int half = lane >> 4;
  const int lr = lane & 15;
  const int m0 = mt * 64;
  const int n0 = nt * 32;

  const float* __restrict__ Arow[4];
#pragma unroll
  for (int i = 0; i < 4; ++i) {
    int r = m0 + i * 16 + lr;
    if (r >= M) r = M - 1;                      // clamp; garbage rows unstored
    Arow[i] = A + (size_t)r * K;
  }
  int col[2];
#pragma unroll
  for (int j = 0; j < 2; ++j) col[j] = n0 + j * 16 + lr;

  v8f c[4][2] = {};
  for (int k = 0; k < K; k += 4) {
    const int ka = k + 2 * half;
    v2f a[4], b[2];
#pragma unroll
    for (int i = 0; i < 4; ++i) {
      a[i][0] = Arow[i][ka];
      a[i][1] = Arow[i][ka + 1];
    }
#pragma unroll
    for (int j = 0; j < 2; ++j) {
      b[j][0] = W[(size_t)ka * N + col[j]];
      b[j][1] = W[(size_t)(ka + 1) * N + col[j]];
    }
#pragma unroll
    for (int i = 0; i < 4; ++i)
#pragma unroll
      for (int j = 0; j < 2; ++j)
        c[i][j] = __builtin_amdgcn_wmma_f32_16x16x4_f32(
            false, a[i], false, b[j], (short)0, c[i][j], false, false);
  }

#pragma unroll
  for (int j = 0; j < 2; ++j) {
    const float bv = bias ? bias[col[j]] : 0.0f;
#pragma unroll
    for (int i = 0; i < 4; ++i) {
      const int baseRow = m0 + i * 16 + half * 8;
#pragma unroll
      for (int r = 0; r < 8; ++r) {
        const int row = baseRow + r;
        if (row < M) {
          float v = c[i][j][r] + bv;
          if (relu) v = fmaxf(v, 0.0f);
          C[(size_t)row * N + col[j]] = v;
        }
      }
    }
  }
}

// ---------------------------------------------------------------------------
// Simple 16x16-per-wave WMMA GEMM fallback (N % 16 == 0).
// ---------------------------------------------------------------------------
__global__ __launch_bounds__(128, 1) void gemm_wmma_f32_16x16(
    const float* __restrict__ A, const float* __restrict__ W,
    const float* __restrict__ bias, float* __restrict__ C,
    int M, int K, int N, int relu) {
  const int lane = threadIdx.x & 31;
  const int wave = threadIdx.x >> 5;
  const int ntc = N >> 4;
  const int mtc = (M + 15) >> 4;
  const long tile = (long)blockIdx.x * 4 + wave;
  if (tile >= (long)mtc * ntc) return;
  const int mt = (int)(tile / ntc);
  const int nt = (int)(tile % ntc);

  const int half = lane >> 4;
  const int lr = lane & 15;
  int rowA = mt * 16 + lr;
  if (rowA >= M) rowA = M - 1;
  const float* __restrict__ Arow = A + (size_t)rowA * K;
  const int colB = nt * 16 + lr;

  v8f c = {};
  for (int k = 0; k < K; k += 4) {
    const int ka = k + 2 * half;
    v2f a, b;
    a[0] = Arow[ka];
    a[1] = Arow[ka + 1];
    b[0] = W[(size_t)ka * N + colB];
    b[1] = W[(size_t)(ka + 1) * N + colB];
    c = __builtin_amdgcn_wmma_f32_16x16x4_f32(false, a, false, b,
                                              (short)0, c, false, false);
  }
  const float bv = bias ? bias[colB] : 0.0f;
  const int baseRow = mt * 16 + half * 8;
#pragma unroll
  for (int r = 0; r < 8; ++r) {
    const int row = baseRow + r;
    if (row < M) {
      float v = c[r] + bv;
      if (relu) v = fmaxf(v, 0.0f);
      C[(size_t)row * N + colB] = v;
    }
  }
}

// ---------------------------------------------------------------------------
// Bilinear helpers (match reference: clip indices, mask taps)
// ---------------------------------------------------------------------------
__device__ __forceinline__ float tap_(const float* __restrict__ base, int xi,
                                      int yi, int H, int Wd, int stride) {
  const bool ok = (xi >= 0) && (xi < Wd) && (yi >= 0) && (yi < H);
  const int xc = min(max(xi, 0), Wd - 1);
  const int yc = min(max(yi, 0), H - 1);
  const float v = base[(size_t)(yc * Wd + xc) * stride];
  return ok ? v : 0.0f;
}

__device__ __forceinline__ float bilin_(const float* __restrict__ base,
                                        float lx, float ly, int H, int Wd,
                                        int stride) {
  const float x = lx * Wd - 0.5f;
  const float y = ly * H - 0.5f;
  const float x0f = floorf(x), y0f = floorf(y);
  const float wx = x - x0f, wy = y - y0f;
  const int x0 = (int)x0f, y0 = (int)y0f;
  const float v00 = tap_(base, x0, y0, H, Wd, stride);
  const float v10 = tap_(base, x0 + 1, y0, H, Wd, stride);
  const float v01 = tap_(base, x0, y0 + 1, H, Wd, stride);
  const float v11 = tap_(base, x0 + 1, y0 + 1, H, Wd, stride);
  return v00 * (1.f - wx) * (1.f - wy) + v10 * wx * (1.f - wy) +
         v01 * (1.f - wx) * wy + v11 * wx * wy;
}

// ---------------------------------------------------------------------------
// init: pos embedding, queries = queries_prev + bev_q_init, qpos = queries+pos
// ---------------------------------------------------------------------------
__global__ void init_kernel(const float* __restrict__ qprev,
                            const float* __restrict__ qinit,
                            float* __restrict__ pos,
                            float* __restrict__ queries,
                            float* __restrict__ qpos) {
  const int t = blockIdx.x * 256 + threadIdx.x;
  if (t >= NQ * DM) return;
  const int q = t >> 8;
  const int d = t & 255;
  const int row = q / BEV_W;
  const int col = q % BEV_W;
  const float scale = 6.283185307179586f;
  float pv;
  if (d < 128) {
    const int i = d >> 1;
    const float v = (row + 0.5f) / (100.0f + 1e-6f) * scale;
    const float tdim = powf(10000.0f, (float)i / 64.0f);
    const float p = v / tdim;
    pv = (d & 1) ? cosf(p) : sinf(p);
  } else {
    const int dd = d - 128;
    const int i = dd >> 1;
    const float v = (col + 0.5f) / (100.0f + 1e-6f) * scale;
    const float tdim = powf(10000.0f, (float)i / 64.0f);
    const float p = v / tdim;
    pv = (dd & 1) ? cosf(p) : sinf(p);
  }
  const float qv = qprev[t] + qinit[t];
  pos[t] = pv;
  queries[t] = qv;
  qpos[t] = qv + pv;
}

// ---------------------------------------------------------------------------
// Projection: ref3d (NCAM,NQ,NZ,2) and valid (NCAM,NQ)
// ---------------------------------------------------------------------------
__global__ void project_kernel(const float* __restrict__ I,
                               const float* __restrict__ E,
                               float* __restrict__ ref,
                               float* __restrict__ valid) {
  const int t = blockIdx.x * 256 + threadIdx.x;
  if (t >= NCAM * NQ) return;
  const int n = t / NQ;
  const int q = t % NQ;
  const int row = q / BEV_W;
  const int col = q % BEV_W;
  const float a = 50.0f - (row + 0.5f);
  const float b = 50.0f - (col + 0.5f);
  const float* En = E + n * 16;
  const float* In = I + n * 9;
  bool anyv = false;
  for (int zi = 0; zi < NZC; ++zi) {
    const float zc = (float)zi * 2.0f;
    float cam[3];
#pragma unroll
    for (int i = 0; i < 3; ++i)
      cam[i] = En[i * 4 + 0] * a + En[i * 4 + 1] * b + En[i * 4 + 2] * zc +
               En[i * 4 + 3];
    float img[3];
#pragma unroll
    for (int i = 0; i < 3; ++i)
      img[i] = In[i * 3 + 0] * cam[0] + In[i * 3 + 1] * cam[1] +
               In[i * 3 + 2] * cam[2];
    const float z = img[2];
    const float zz = fmaxf(z, 1e-5f);
    const float px = img[0] / zz;
    const float py = img[1] / zz;
    const bool m = (z > 1e-5f) && (px > 0.f) && (px < 800.f) && (py > 0.f) &&
                   (py < 450.f);
    anyv = anyv || m;
    ref[((size_t)t * NZC + zi) * 2 + 0] = px / 800.0f;
    ref[((size_t)t * NZC + zi) * 2 + 1] = py / 450.0f;
  }
  valid[t] = anyv ? 1.0f : 0.0f;
}

// ---------------------------------------------------------------------------
// In-place softmax over contiguous groups of gsz (gsz <= 32)
// ---------------------------------------------------------------------------
__global__ void softmax_groups(float* __restrict__ a, int ngroups, int gsz) {
  const int i = blockIdx.x * 256 + threadIdx.x;
  if (i >= ngroups) return;
  float* p = a + (size_t)i * gsz;
  float t[32];
  float m = -1e30f;
  for (int j = 0; j < gsz; ++j) {
    t[j] = p[j];
    m = fmaxf(m, t[j]);
  }
  float s = 0.f;
  for (int j = 0; j < gsz; ++j) {
    t[j] = expf(t[j] - m);
    s += t[j];
  }
  const float inv = 1.0f / s;
  for (int j = 0; j < gsz; ++j) p[j] = t[j] * inv;
}

// ---------------------------------------------------------------------------
// Self deformable sampling: wave per (q,h), lane = channel d
// ---------------------------------------------------------------------------
__global__ void self_sample(const float* __restrict__ val,
                            const float* __restrict__ off,
                            const float* __restrict__ aw,
                            float* __restrict__ sout) {
  const int gw = blockIdx.x * 8 + (threadIdx.x >> 5);
  if (gw >= NQ * NH) return;
  const int d = threadIdx.x & 31;
  const int h = gw % NH;
  const int q = gw / NH;
  const int row = q / BEV_W;
  const int col = q % BEV_W;
  const float rx = (float)col * (1.0f / 99.0f);
  const float ry = (float)row * (1.0f / 99.0f);
  const float* vimg = val + h * HD + d;
  float s = 0.f;
#pragma unroll
  for (int p = 0; p < NPT; ++p) {
    const float lx = rx + off[(((size_t)q * NH + h) * NPT + p) * 2 + 0] * 0.01f;
    const float ly = ry + off[(((size_t)q * NH + h) * NPT + p) * 2 + 1] * 0.01f;
    const float w = aw[((size_t)q * NH + h) * NPT + p];
    s += w * bilin_(vimg, lx, ly, BEV_H, BEV_W, DM);
  }
  sout[(size_t)q * DM + h * HD + d] = s;
}

// ---------------------------------------------------------------------------
// Build per-level feature rows: A[n*HW+s][d] = feat[n][d][s] + lev[d] + cam[n][d]
// ---------------------------------------------------------------------------
__global__ void build_f(const float* __restrict__ feat,
                        const float* __restrict__ lev,
                        const float* __restrict__ cams,
                        float* __restrict__ A, int HW, int total) {
  const int t = blockIdx.x * 256 + threadIdx.x;
  if (t >= total) return;
  const int d = t & 255;
  const int rest = t >> 8;
  const int s = rest % HW;
  const int n = rest / HW;
  A[t] = feat[((size_t)(n * DM + d)) * HW + s] + lev[d] + cams[n * DM + d];
}

// ---------------------------------------------------------------------------
// Cross deformable sampling: wave per (n,q,h), lane = channel d
// ---------------------------------------------------------------------------
__global__ void cross_sample(const float* __restrict__ v,
                             const float* __restrict__ offc,
                             const float* __restrict__ awc,
                             const float* __restrict__ ref,
                             float* __restrict__ acc) {
  const int gw = blockIdx.x * 8 + (threadIdx.x >> 5);
  if (gw >= NCAM * NQ * NH) return;
  const int d = threadIdx.x & 31;
  const int h = gw % NH;
  const int rest = gw / NH;
  const int q = rest % NQ;
  const int n = rest / NQ;
  const int Hs[4] = {56, 28, 14, 7};
  const int Ws[4] = {100, 50, 25, 13};
  const int HWs[4] = {5600, 1400, 350, 91};
  const int vb[4] = {0, 33600, 42000, 44100};
  const float* refq = ref + ((size_t)(n * NQ + q)) * (NZC * 2);
  const float* offr = offc + (size_t)q * 512 + h * 64;  // (l,z,p,2)
  const float* awr = awc + (size_t)q * 256 + h * 32;    // (l,z,p)
  float s = 0.f;
  for (int l = 0; l < NLV; ++l) {
    const int Hl = Hs[l], Wl = Ws[l], HW = HWs[l];
    const float* vimg = v + ((size_t)(vb[l] + n * HW)) * DM + h * HD + d;
    const float invW = 1.0f / (float)Wl;
    const float invH = 1.0f / (float)Hl;
#pragma unroll
    for (int z = 0; z < NZC; ++z) {
      const float rx = refq[z * 2 + 0];
      const float ry = refq[z * 2 + 1];
#pragma unroll
      for (int p = 0; p < NPT; ++p) {
        const int j = (l * NZC + z) * NPT + p;
        const float lx = rx + offr[j * 2 + 0] * invW;
        const float ly = ry + offr[j * 2 + 1] * invH;
        s += awr[j] * bilin_(vimg, lx, ly, Hl, Wl, DM);
      }
    }
  }
  acc[((size_t)(n * NQ + q)) * DM + h * HD + d] = s;
}

// ---------------------------------------------------------------------------
// LayerNorm over DM=256 (block per row), with residual; optional qpos output
// ---------------------------------------------------------------------------
__device__ __forceinline__ float ln_body(float v, const float* __restrict__ g,
                                         const float* __restrict__ be, int d,
                                         float* sm) {
  sm[d] = v;
  __syncthreads();
  for (int s = 128; s > 0; s >>= 1) {
    if (d < s) sm[d] += sm[d + s];
    __syncthreads();
  }
  const float mu = sm[0] * (1.0f / 256.0f);
  __syncthreads();
  const float dv = v - mu;
  sm[d] = dv * dv;
  __syncthreads();
  for (int s = 128; s > 0; s >>= 1) {
    if (d < s) sm[d] += sm[d + s];
    __syncthreads();
  }
  const float var = sm[0] * (1.0f / 256.0f);
  __syncthreads();
  return dv * rsqrtf(var + 1e-5f) * g[d] + be[d];
}

__global__ void ln_res_kernel(const float* __restrict__ x,
                              const float* __restrict__ res,
                              const float* __restrict__ g,
                              const float* __restrict__ be,
                              float* __restrict__ out,
                              float* __restrict__ outpos,
                              const float* __restrict__ pos) {
  __shared__ float sm[256];
  const int q = blockIdx.x;
  const int d = threadIdx.x;
  const size_t idx = (size_t)q * DM + d;
  const float v = x[idx] + res[idx];
  const float y = ln_body(v, g, be, d, sm);
  out[idx] = y;
  if (outpos) outpos[idx] = y + pos[idx];
}

// ---------------------------------------------------------------------------
// Camera-masked mean + residual + LN
// ---------------------------------------------------------------------------
__global__ void camred_ln(const float* __restrict__ outc,
                          const float* __restrict__ valid,
                          const float* __restrict__ q2,
                          const float* __restrict__ g,
                          const float* __restrict__ be,
                          float* __restrict__ out) {
  __shared__ float sm[256];
  const int q = blockIdx.x;
  const int d = threadIdx.x;
  float cnt = 0.f, s = 0.f;
#pragma unroll
  for (int n = 0; n < NCAM; ++n) {
    const float vl = valid[(size_t)n * NQ + q];
    cnt += vl;
    s += outc[((size_t)n * NQ + q) * DM + d] * vl;
  }
  cnt = fmaxf(cnt, 1.0f);
  const float v = q2[(size_t)q * DM + d] + s / cnt;
  out[(size_t)q * DM + d] = ln_body(v, g, be, d, sm);
}

// ---------------------------------------------------------------------------
static inline void gemm(const float* A, const float* W, const float* bias,
                        float* C, int M, int K, int N, int relu,
                        hipStream_t s) {
  if ((N & 31) == 0) {
    const long mt = (M + 63) / 64;
    const long nt = N / 32;
    const long tiles = mt * nt;
    const int blocks = (int)((tiles + 7) / 8);
    gemm_wmma_f32_64x32<<<blocks, 256, 0, s>>>(A, W, bias, C, M, K, N, relu);
  } else {
    const long mt = (M + 15) / 16;
    const long nt = N / 16;
    const long tiles = mt * nt;
    const int blocks = (int)((tiles + 3) / 4);
    gemm_wmma_f32_16x16<<<blocks, 128, 0, s>>>(A, W, bias, C, M, K, N, relu);
  }
}

extern "C" void kernel_launch(void* const* d_in, const int* in_sizes, int n_in,
                              void* d_out, int out_size, void* d_ws,
                              size_t ws_size, hipStream_t stream) {
  const float* queries_prev = (const float*)d_in[0];
  const float* features[4] = {(const float*)d_in[1], (const float*)d_in[2],
                              (const float*)d_in[3], (const float*)d_in[4]};
  const float* intr = (const float*)d_in[5];
  const float* extr = (const float*)d_in[6];
  const float* bev_q_init = (const float*)d_in[7];
  const float* level_embed = (const float*)d_in[8];
  const float* cams_embed = (const float*)d_in[9];
  const float* ws_off = (const float*)d_in[10];
  const float* bs_off = (const float*)d_in[11];
  const float* ws_attn = (const float*)d_in[12];
  const float* bs_attn = (const float*)d_in[13];
  const float* ws_val = (const float*)d_in[14];
  const float* bs_val = (const float*)d_in[15];
  const float* ws_out = (const float*)d_in[16];
  const float* bs_out = (const float*)d_in[17];
  const float* g1 = (const float*)d_in[18];
  const float* be1 = (const float*)d_in[19];
  const float* wc_off = (const float*)d_in[20];
  const float* bc_off = (const float*)d_in[21];
  const float* wc_attn = (const float*)d_in[22];
  const float* bc_attn = (const float*)d_in[23];
  const float* wc_val = (const float*)d_in[24];
  const float* bc_val = (const float*)d_in[25];
  const float* wc_out = (const float*)d_in[26];
  const float* bc_out = (const float*)d_in[27];
  const float* g2 = (const float*)d_in[28];
  const float* be2 = (const float*)d_in[29];
  const float* w1 = (const float*)d_in[30];
  const float* b1 = (const float*)d_in[31];
  const float* w2 = (const float*)d_in[32];
  const float* b2 = (const float*)d_in[33];
  const float* g3 = (const float*)d_in[34];
  const float* be3 = (const float*)d_in[35];

  // ---- workspace layout (floats) ----
  float* ws = (float*)d_ws;
  size_t o = 0;
  float* pos = ws + o;      o += 2560000;
  float* queries = ws + o;  o += 2560000;
  float* qpos = ws + o;     o += 2560000;
  float* R1 = ws + o;       o += 8640000;  // off|aw|val|sout|sa , later offc|awc
  float* off = R1;
  float* aw = R1 + 640000;
  float* val = R1 + 960000;
  float* sout = R1 + 3520000;
  float* sa = R1 + 6080000;
  float* offc = R1;              // 5,120,000 (after self-attn stage)
  float* awc = R1 + 5120000;     // 2,560,000
  float* q2 = ws + o;       o += 2560000;
  float* qpos2 = ws + o;    o += 2560000;
  float* R2 = ws + o;       o += 8601600;  // Abuf, later queries3
  float* Abuf = R2;
  float* queries3 = R2;
  float* vbuf = ws + o;     o += 11429376;  // 44646 rows x 256
  float* ref = ws + o;      o += 240000;
  float* valid = ws + o;    o += 60000;
  float* R3 = ws + o;       o += 15360000;  // acc, later h1
  float* acc = R3;
  float* h1 = R3;
  float* R4 = ws + o;       o += 15360000;  // outc, later h2
  float* outc = R4;
  float* h2 = R4;
  (void)ws_size; (void)in_sizes; (void)n_in; (void)out_size;

  const int HWs[4] = {5600, 1400, 350, 91};
  const int vb[4] = {0, 33600, 42000, 44100};

  // 1. pos / queries / qpos
  init_kernel<<<(NQ * DM + 255) / 256, 256, 0, stream>>>(
      queries_prev, bev_q_init, pos, queries, qpos);
  // 2. projection
  project_kernel<<<(NCAM * NQ + 255) / 256, 256, 0, stream>>>(intr, extr, ref,
                                                              valid);
  // 3-5. self-attn projections
  gemm(qpos, ws_off, bs_off, off, NQ, DM, NH * NPT * 2, 0, stream);
  gemm(qpos, ws_attn, bs_attn, aw, NQ, DM, NH * NPT, 0, stream);
  softmax_groups<<<(NQ * NH + 255) / 256, 256, 0, stream>>>(aw, NQ * NH, NPT);
  gemm(queries, ws_val, bs_val, val, NQ, DM, DM, 0, stream);
  // 6. self deformable sampling
  self_sample<<<(NQ * NH + 7) / 8, 256, 0, stream>>>(val, off, aw, sout);
  // 7. output projection + residual LN (also q2+pos)
  gemm(sout, ws_out, bs_out, sa, NQ, DM, DM, 0, stream);
  ln_res_kernel<<<NQ, 256, 0, stream>>>(queries, sa, g1, be1, q2, qpos2, pos);
  // 8-9. cross-attn projections
  gemm(qpos2, wc_off, bc_off, offc, NQ, DM, NH * NLV * NZC * NPT * 2, 0,
       stream);
  gemm(qpos2, wc_attn, bc_attn, awc, NQ, DM, NH * NLV * NZC * NPT, 0, stream);
  softmax_groups<<<(NQ * NH + 255) / 256, 256, 0, stream>>>(awc, NQ * NH,
                                                            NLV * NZC * NPT);
  // 10. per-level value projections
  for (int l = 0; l < NLV; ++l) {
    const int HW = HWs[l];
    const int total = NCAM * HW * DM;
    build_f<<<(total + 255) / 256, 256, 0, stream>>>(
        features[l], level_embed + l * DM, cams_embed, Abuf, HW, total);
    gemm(Abuf, wc_val, bc_val, vbuf + (size_t)vb[l] * DM, NCAM * HW, DM, DM, 0,
         stream);
  }
  // 11. cross deformable sampling
  cross_sample<<<(NCAM * NQ * NH + 7) / 8, 256, 0, stream>>>(vbuf, offc, awc,
                                                             ref, acc);
  // 12. cross output projection
  gemm(acc, wc_out, bc_out, outc, NCAM * NQ, DM, DM, 0, stream);
  // 13. camera-masked mean + residual LN
  camred_ln<<<NQ, 256, 0, stream>>>(outc, valid, q2, g2, be2, queries3);
  // 14. FFN
  gemm(queries3, w1, b1, h1, NQ, DM, DFF, 1, stream);
  gemm(h1, w2, b2, h2, NQ, DFF, DM, 0, stream);
  // 15. final residual LN -> output
  ln_res_kernel<<<NQ, 256, 0, stream>>>(queries3, h2, g3, be3, (float*)d_out,
                                        nullptr, nullptr);
}